// HetGAT_49065706389922
// MI455X (gfx1250) — compile-verified
//
#include <hip/hip_runtime.h>

#define NODES  50000
#define EDGES  1600000
#define EMB    128
#define NHEADS 8
#define HDIM   16

typedef float v2f __attribute__((ext_vector_type(2)));
typedef float v8f __attribute__((ext_vector_type(8)));

// ---- monotonic float<->uint key (so u32 atomicMax == float max) ----
__device__ __forceinline__ unsigned fkey(float f) {
  unsigned u = __float_as_uint(f);
  return (u & 0x80000000u) ? ~u : (u | 0x80000000u);
}
__device__ __forceinline__ float funkey(unsigned u) {
  return __uint_as_float((u & 0x80000000u) ? (u & 0x7FFFFFFFu) : ~u);
}
// fkey(-FLT_MAX) == 0x00800000u  (init sentinel for segment max)

// ---- projection GEMM: C[N,128] = A[N,128] @ B[128,128], f32 WMMA ----
// one wave -> one 16x16 tile, K loop of 32 x v_wmma_f32_16x16x4_f32
__global__ void gemm_wmma_k(const float* __restrict__ A, const float* __restrict__ B,
                            float* __restrict__ C) {
  int wave = (blockIdx.x * blockDim.x + threadIdx.x) >> 5;
  int lane = threadIdx.x & 31;
  int mt = wave >> 3;       // 8 column tiles (128/16)
  int nt = wave & 7;
  if (mt * 16 >= NODES) return;   // wave-uniform; never taken (50000 % 16 == 0)
  int l16  = lane & 15;
  int half = lane >> 4;
  const float* arow = A + (size_t)(mt * 16 + l16) * EMB;
  int col = nt * 16 + l16;
  v8f acc = {};
  for (int k0 = 0; k0 < EMB; k0 += 4) {
    // A 16x4: VGPR0 = K0(lanes0-15)/K2(lanes16-31); VGPR1 = K1/K3 -> contiguous pair
    v2f a = *(const v2f*)(arow + k0 + 2 * half);
    v2f b;
    b.x = B[(size_t)(k0 + 2 * half + 0) * EMB + col];
    b.y = B[(size_t)(k0 + 2 * half + 1) * EMB + col];
    acc = __builtin_amdgcn_wmma_f32_16x16x4_f32(false, a, false, b, (short)0, acc,
                                                false, false);
  }
  // D: VGPR v -> row (v + 8*half), col = l16
  float* crow = C + (size_t)(mt * 16 + 8 * half) * EMB + nt * 16 + l16;
#pragma unroll
  for (int v = 0; v < 8; ++v) crow[(size_t)v * EMB] = acc[v];
}

// ---- el/er = einsum('nhd,hd->nh', feat, a) ----
__global__ void node_attn_k(const float* __restrict__ feat,
                            const float* __restrict__ al, const float* __restrict__ ar,
                            float* __restrict__ el, float* __restrict__ er) {
  int i = blockIdx.x * blockDim.x + threadIdx.x;
  if (i >= NODES * NHEADS) return;
  int n = i >> 3, h = i & 7;
  const float* f = feat + (size_t)n * EMB + h * HDIM;
  const float* a = al + h * HDIM;
  const float* b = ar + h * HDIM;
  float sl = 0.f, sr = 0.f;
#pragma unroll
  for (int d = 0; d < HDIM; ++d) { float v = f[d]; sl += v * a[d]; sr += v * b[d]; }
  el[i] = sl;
  er[i] = sr;
}

// ---- init accumulators: rst=0, emax=key(-FLT_MAX), esum=0 ----
__global__ void init_k(unsigned* __restrict__ emax_u, float* __restrict__ esum,
                       float* __restrict__ rst) {
  int i = blockIdx.x * blockDim.x + threadIdx.x;
  if (i < NODES * EMB) rst[i] = 0.f;
  if (i < NODES * NHEADS) { emax_u[i] = 0x00800000u; esum[i] = 0.f; }
}

// ---- pass 1: segment max of leaky_relu(el[src]+er[dst]) via u32 atomicMax ----
__global__ void edge_max_k(const int* __restrict__ src, const int* __restrict__ dst,
                           const float* __restrict__ el, const float* __restrict__ er,
                           unsigned* __restrict__ emax_u) {
  int e = blockIdx.x * blockDim.x + threadIdx.x;
  if (e >= EDGES) return;
  int s = src[e], d = dst[e];
  float vl[8], vr[8];
  *(float4*)&vl[0] = ((const float4*)(el + (size_t)s * 8))[0];
  *(float4*)&vl[4] = ((const float4*)(el + (size_t)s * 8))[1];
  *(float4*)&vr[0] = ((const float4*)(er + (size_t)d * 8))[0];
  *(float4*)&vr[4] = ((const float4*)(er + (size_t)d * 8))[1];
#pragma unroll
  for (int h = 0; h < 8; ++h) {
    float x = vl[h] + vr[h];
    x = x > 0.f ? x : 0.2f * x;
    atomicMax(&emax_u[(size_t)d * 8 + h], fkey(x));
  }
}

// ---- decode keys back to float in place; zero-in-degree -> 0 (ref semantics) ----
__global__ void decode_k(unsigned* __restrict__ emax_u) {
  int i = blockIdx.x * blockDim.x + threadIdx.x;
  if (i >= NODES * NHEADS) return;
  float v = funkey(emax_u[i]);
  ((float*)emax_u)[i] = (v < -3.0e38f) ? 0.f : v;
}

// ---- pass 2: esum[dst,h] += exp(e - emax[dst,h]) ----
__global__ void edge_sum_k(const int* __restrict__ src, const int* __restrict__ dst,
                           const float* __restrict__ el, const float* __restrict__ er,
                           const float* __restrict__ emax, float* __restrict__ esum) {
  int e = blockIdx.x * blockDim.x + threadIdx.x;
  if (e >= EDGES) return;
  int s = src[e], d = dst[e];
  float vl[8], vr[8], vm[8];
  *(float4*)&vl[0] = ((const float4*)(el + (size_t)s * 8))[0];
  *(float4*)&vl[4] = ((const float4*)(el + (size_t)s * 8))[1];
  *(float4*)&vr[0] = ((const float4*)(er + (size_t)d * 8))[0];
  *(float4*)&vr[4] = ((const float4*)(er + (size_t)d * 8))[1];
  *(float4*)&vm[0] = ((const float4*)(emax + (size_t)d * 8))[0];
  *(float4*)&vm[4] = ((const float4*)(emax + (size_t)d * 8))[1];
#pragma unroll
  for (int h = 0; h < 8; ++h) {
    float x = vl[h] + vr[h];
    x = x > 0.f ? x : 0.2f * x;
    atomicAdd(&esum[(size_t)d * 8 + h], __expf(x - vm[h]));
  }
}

// ---- pass 3: alpha + rst[dst] += alpha * feat[src]; thread per (edge, head) ----
__global__ void edge_aggr_k(const int* __restrict__ src, const int* __restrict__ dst,
                            const float* __restrict__ el, const float* __restrict__ er,
                            const float* __restrict__ emax, const float* __restrict__ esum,
                            const float* __restrict__ feat, float* __restrict__ rst,
                            float* __restrict__ alpha_out) {
  int i = blockIdx.x * blockDim.x + threadIdx.x;
  if (i >= EDGES * NHEADS) return;
  int e = i >> 3, h = i & 7;
  int s = src[e], d = dst[e];
  int dh = d * NHEADS + h;
  float x = el[(size_t)s * NHEADS + h] + er[dh];
  x = x > 0.f ? x : 0.2f * x;
  float a = __expf(x - emax[dh]) / fmaxf(esum[dh], 1e-9f);
  if (alpha_out) alpha_out[(size_t)e * NHEADS + h] = a;
  const float4* f = (const float4*)(feat + (size_t)s * EMB + h * HDIM);
  float* r = rst + (size_t)d * EMB + h * HDIM;
#pragma unroll
  for (int q = 0; q < 4; ++q) {
    float4 v = f[q];
    atomicAdd(r + q * 4 + 0, a * v.x);
    atomicAdd(r + q * 4 + 1, a * v.y);
    atomicAdd(r + q * 4 + 2, a * v.z);
    atomicAdd(r + q * 4 + 3, a * v.w);
  }
}

// ---- out = elu(rst + h_in + bias) ----
__global__ void finalize_k(const float* __restrict__ rst, const float* __restrict__ hin,
                           const float* __restrict__ bias, float* __restrict__ hout) {
  int i = blockIdx.x * blockDim.x + threadIdx.x;
  if (i >= NODES * EMB) return;
  float v = rst[i] + hin[i] + bias[i & (EMB - 1)];
  hout[i] = v > 0.f ? v : (__expf(v) - 1.f);
}

extern "C" void kernel_launch(void* const* d_in, const int* in_sizes, int n_in,
                              void* d_out, int out_size, void* d_ws, size_t ws_size,
                              hipStream_t stream) {
  const float* fs    = (const float*)d_in[0];  // [2, N, 128]
  const float* Wall  = (const float*)d_in[1];  // [2, 2, 128, 128]
  const float* alall = (const float*)d_in[2];  // [2, 2, 8, 16]
  const float* arall = (const float*)d_in[3];  // [2, 2, 8, 16]
  const float* ball  = (const float*)d_in[4];  // [2, 2, 128]
  const int*   edges = (const int*)d_in[5];    // [2, 2, E]
  float* out = (float*)d_out;                  // [2,N,128] ++ [2,E,8]

  const size_t NE  = (size_t)NODES * EMB;      // 6,400,000
  const size_t NHt = (size_t)NODES * NHEADS;   //   400,000
  const size_t EH  = (size_t)EDGES * NHEADS;   // 12,800,000

  char* ws = (char*)d_ws;
  float* h_buf = (float*)(ws);                              // 25.6 MB
  float* feat  = (float*)(ws + 25600000);                   // 25.6 MB
  float* rst   = (float*)(ws + 51200000);                   // 25.6 MB
  float* el    = (float*)(ws + 76800000);                   //  1.6 MB
  float* er    = (float*)(ws + 78400000);                   //  1.6 MB
  float* emax  = (float*)(ws + 80000000);                   //  1.6 MB
  float* esum  = (float*)(ws + 81600000);                   //  1.6 MB (total 83.2 MB)

  const int T = 256;
  const int gGemm = (NODES / 16) * 8 * 32 / T;  // 3125 blocks, 1 wave = 1 tile
  const int gNE   = (int)((NE + T - 1) / T);
  const int gNH   = (int)((NHt + T - 1) / T);
  const int gE    = (EDGES + T - 1) / T;
  const int gEH   = (int)((EH + T - 1) / T);

  for (int p = 0; p < 2; ++p) {
    const int* src = edges + (size_t)p * 2 * EDGES;
    const int* dst = src + EDGES;
    for (int l = 0; l < 2; ++l) {
      const int pl = p * 2 + l;
      const float* hin = (l == 0) ? (fs + (size_t)p * NE) : h_buf;
      const float* Wpl = Wall + (size_t)pl * EMB * EMB;
      const float* al  = alall + (size_t)pl * NHEADS * HDIM;
      const float* ar  = arall + (size_t)pl * NHEADS * HDIM;
      const float* bpl = ball + (size_t)pl * EMB;
      const bool last = (l == 1);
      float* hout      = last ? (out + (size_t)p * NE) : h_buf;
      float* alpha_out = last ? (out + 2 * NE + (size_t)p * EH) : (float*)nullptr;

      gemm_wmma_k<<<gGemm, T, 0, stream>>>(hin, Wpl, feat);
      node_attn_k<<<gNH, T, 0, stream>>>(feat, al, ar, el, er);
      init_k<<<gNE, T, 0, stream>>>((unsigned*)emax, esum, rst);
      edge_max_k<<<gE, T, 0, stream>>>(src, dst, el, er, (unsigned*)emax);
      decode_k<<<gNH, T, 0, stream>>>((unsigned*)emax);
      edge_sum_k<<<gE, T, 0, stream>>>(src, dst, el, er, emax, esum);
      edge_aggr_k<<<gEH, T, 0, stream>>>(src, dst, el, er, emax, esum, feat, rst, alpha_out);
      finalize_k<<<gNE, T, 0, stream>>>(rst, hin, bpl, hout);
    }
  }
}